// Embedding_61108794688267
// MI455X (gfx1250) — compile-verified
//
#include <hip/hip_runtime.h>

// Embedding row-gather: out[token, :] = w[x[token], :]
//   x: (8*2048,) int32        -> d_in[0]
//   w: (50257, 512) float32   -> d_in[1]
//   out: (8*2048, 512) float32 -> d_out
//
// Pure bandwidth problem: ~32 MB gathered reads (table fits in 192 MB L2)
// + 32 MB streaming writes. Target: wide b128 vmem ops, NT stores for the
// write-once output stream so it doesn't evict the embedding table from L2.
// Roofline at 23.3 TB/s HBM: ~64 MB / 23.3 TB/s ~= 2.7 us.

#define EMBED 512            // floats per row
#define VEC   4              // floats per lane (b128)
#define COLS4 (EMBED / VEC)  // 128 16-byte chunks per row

// Native clang vector type: accepted by __builtin_nontemporal_store and
// lowers to global_load_b128 / global_store_b128.
typedef float v4f __attribute__((ext_vector_type(4)));

__global__ __launch_bounds__(256) void
embedding_gather_kernel(const int* __restrict__ x,
                        const float* __restrict__ w,
                        float* __restrict__ out,
                        int n_tokens)
{
    // Flat index over (token, col4): each thread moves one 16-byte chunk.
    const long gid   = (long)blockIdx.x * blockDim.x + threadIdx.x;
    const long total = (long)n_tokens * COLS4;
    if (gid >= total) return;

    const int token = (int)(gid >> 7);   // gid / COLS4  (COLS4 == 128)
    const int col4  = (int)(gid & (COLS4 - 1));

    // Uniform across each wave32 (128 chunks per row > 32 lanes), so this
    // coalesces into a single cacheline request per wave and hits L2.
    const int idx = x[token];

    // 16-byte aligned: rows are 2048 B, col4 steps are 16 B.
    const v4f* __restrict__ src = (const v4f*)(w + (long)idx * EMBED) + col4;
    v4f v = *src;   // regular-temporal: keep hot vocab rows resident in L2

    v4f* __restrict__ dst = (v4f*)(out + (long)token * EMBED) + col4;
    // Non-temporal store: 32 MB write-once stream, don't pollute L2.
    __builtin_nontemporal_store(v, dst);
}

extern "C" void kernel_launch(void* const* d_in, const int* in_sizes, int n_in,
                              void* d_out, int out_size, void* d_ws, size_t ws_size,
                              hipStream_t stream)
{
    const int*   x = (const int*)d_in[0];    // (8, 2048) int32 indices
    const float* w = (const float*)d_in[1];  // (50257, 512) float32 table
    float*     out = (float*)d_out;          // (8, 2048, 512) float32

    const int n_tokens = in_sizes[0];                // 16384
    const long total   = (long)n_tokens * COLS4;     // 16-byte chunks
    const int  block   = 256;                        // 8 wave32 fronts
    const int  grid    = (int)((total + block - 1) / block);

    embedding_gather_kernel<<<grid, block, 0, stream>>>(x, w, out, n_tokens);
}